// EigenSpace_60309930770489
// MI455X (gfx1250) — compile-verified
//
#include <hip/hip_runtime.h>

typedef __attribute__((ext_vector_type(2))) float v2f;
typedef __attribute__((ext_vector_type(4))) float v4f;
typedef __attribute__((ext_vector_type(8))) float v8f;

#define DIM        512             // K (d)
#define BATCH      32              // N (b)
#define M_TOTAL    (512 * 16 * 16) // 131072 rows (c,r,s)
#define TILES_PER_WAVE 2           // two 16-row M tiles per wave
#define ROWS_PER_WAVE  (16 * TILES_PER_WAVE)       // 32
#define WAVES_PER_BLK  8
#define ROWS_PER_BLK   (ROWS_PER_WAVE * WAVES_PER_BLK) // 256
#define FACTOR     1.0f

// out[b*M_TOTAL + m] = sum_d U[m*DIM + d] * style[b*DIM + d] * L[d] + mu[m]
__global__ __launch_bounds__(256) void eigenspace_wmma_kernel(
    const float* __restrict__ style,  // (32, 512)
    const float* __restrict__ U,      // (131072, 512) row-major
    const float* __restrict__ L,      // (512)
    const float* __restrict__ mu,     // (131072)
    float* __restrict__ out)          // (32, 131072)
{
    // Scaled style, pair-interleaved + bank-rotated:
    //   element (b, k) at  (k>>1)*64 + ((2b + 32*((k>>1)&1)) & 63) + (k&1)
    // -> each WMMA B operand {K,K+1} is one aligned ds_load_b64 into an
    //    adjacent VGPR pair (no v_mov repacking), and the rotation puts the
    //    two lane-halves in disjoint bank sets {0..31}/{32..63}:
    //    conflict-free. Exactly 64 KB.
    __shared__ float s_lds[DIM * BATCH];

    const int tid = threadIdx.x;

    // Cooperative fill: consecutive threads read consecutive k -> coalesced
    // global reads of style/L (one-time cost).
    for (int idx = tid; idx < DIM * BATCH; idx += 256) {
        const int b = idx >> 9;          // / DIM
        const int k = idx & (DIM - 1);   // % DIM
        const int p = k >> 1;            // k-pair index
        const int j = k & 1;
        const int rot = (2 * b + ((p & 1) << 5)) & 63;
        s_lds[(p << 6) + rot + j] = style[b * DIM + k] * (L[k] * FACTOR);
    }
    __syncthreads();

    const int wave   = tid >> 5;     // wave32
    const int lane   = tid & 31;
    const int lane16 = lane & 15;
    const int khalf  = lane >> 4;    // 0: lanes 0-15 hold K{0,1}; 1: lanes 16-31 hold K{2,3}

    const int m_base = blockIdx.x * ROWS_PER_BLK + wave * ROWS_PER_WAVE;

    // A-operand streams: one U row per lane per tile (16x4 fp32 A layout:
    // lane = M row, VGPR0/1 = consecutive K pair selected by lane-half).
    const float* Urow0 = U + (size_t)(m_base +      lane16) * DIM + 2 * khalf;
    const float* Urow1 = U + (size_t)(m_base + 16 + lane16) * DIM + 2 * khalf;

    // Per-lane, loop-invariant LDS bases; loop adds k0*32 floats (k0*128 bytes), which
    // folds into DS immediate offsets under unrolling.
    const int rot0  = (2 * lane16 +      (khalf << 5)) & 63;  // batches 0..15
    const int rot1  = (2 * lane16 + 32 + (khalf << 5)) & 63;  // batches 16..31
    const float* sb0 = s_lds + (khalf << 6) + rot0;
    const float* sb1 = s_lds + (khalf << 6) + rot1;

    v8f c00 = {};  // tile0 x batches 0..15
    v8f c01 = {};  // tile0 x batches 16..31
    v8f c10 = {};  // tile1 x batches 0..15
    v8f c11 = {};  // tile1 x batches 16..31

    #pragma unroll 8
    for (int k0 = 0; k0 < DIM; k0 += 4) {
        // A: 8-byte loads, K pair {k0+2h, k0+2h+1}
        v2f a0 = *(const v2f*)(Urow0 + k0);
        v2f a1 = *(const v2f*)(Urow1 + k0);

        // B: one aligned ds_load_b64 per batch group, conflict-free
        v2f b0 = *(const v2f*)(sb0 + k0 * 32);
        v2f b1 = *(const v2f*)(sb1 + k0 * 32);

        // D = A(16x4) x B(4x16) + C, fp32; B operands shared by both M tiles
        c00 = __builtin_amdgcn_wmma_f32_16x16x4_f32(
                  false, a0, false, b0, (short)0, c00, false, false);
        c01 = __builtin_amdgcn_wmma_f32_16x16x4_f32(
                  false, a0, false, b1, (short)0, c01, false, false);
        c10 = __builtin_amdgcn_wmma_f32_16x16x4_f32(
                  false, a1, false, b0, (short)0, c10, false, false);
        c11 = __builtin_amdgcn_wmma_f32_16x16x4_f32(
                  false, a1, false, b1, (short)0, c11, false, false);
    }

    // D layout: VGPR v, lane-half h -> M = v + 8*h, N = lane%16.
    // Each lane's 8 output rows per tile are contiguous in m.
    const int mrow0 = m_base +      8 * khalf;
    const int mrow1 = m_base + 16 + 8 * khalf;

    // mu broadcast over b (same add for both batch groups)
    const float* mup0 = mu + mrow0;
    const float* mup1 = mu + mrow1;
    #pragma unroll
    for (int v = 0; v < 8; ++v) {
        const float mv0 = mup0[v];
        const float mv1 = mup1[v];
        c00[v] += mv0;  c01[v] += mv0;
        c10[v] += mv1;  c11[v] += mv1;
    }

    float* o00 = out + (size_t)lane16        * M_TOTAL + mrow0;
    float* o01 = out + (size_t)(lane16 + 16) * M_TOTAL + mrow0;
    float* o10 = out + (size_t)lane16        * M_TOTAL + mrow1;
    float* o11 = out + (size_t)(lane16 + 16) * M_TOTAL + mrow1;

    *(v4f*)(o00)     = (v4f){c00[0], c00[1], c00[2], c00[3]};
    *(v4f*)(o00 + 4) = (v4f){c00[4], c00[5], c00[6], c00[7]};
    *(v4f*)(o01)     = (v4f){c01[0], c01[1], c01[2], c01[3]};
    *(v4f*)(o01 + 4) = (v4f){c01[4], c01[5], c01[6], c01[7]};
    *(v4f*)(o10)     = (v4f){c10[0], c10[1], c10[2], c10[3]};
    *(v4f*)(o10 + 4) = (v4f){c10[4], c10[5], c10[6], c10[7]};
    *(v4f*)(o11)     = (v4f){c11[0], c11[1], c11[2], c11[3]};
    *(v4f*)(o11 + 4) = (v4f){c11[4], c11[5], c11[6], c11[7]};
}

extern "C" void kernel_launch(void* const* d_in, const int* in_sizes, int n_in,
                              void* d_out, int out_size, void* d_ws, size_t ws_size,
                              hipStream_t stream) {
    const float* style = (const float*)d_in[0]; // (32, 512)
    const float* U     = (const float*)d_in[1]; // (512, 16, 16, 512)
    const float* L     = (const float*)d_in[2]; // (512,)
    const float* mu    = (const float*)d_in[3]; // (512, 16, 16)
    float* out         = (float*)d_out;         // (32, 512, 16, 16)

    (void)in_sizes; (void)n_in; (void)out_size; (void)d_ws; (void)ws_size;

    const int grid = M_TOTAL / ROWS_PER_BLK;    // 512 blocks
    eigenspace_wmma_kernel<<<grid, 256, 0, stream>>>(style, U, L, mu, out);
}